// RelationEncoder_5634997092930
// MI455X (gfx1250) — compile-verified
//
#include <hip/hip_runtime.h>
#include <stdint.h>

// Problem constants (match reference).
#define N_ITEMS   100000
#define EMBED_DIM 64
#define N_REL     3
#define N_EDGES   3200000

#define TILE        256                      // edges staged per block iteration
#define TILE_BYTES  (TILE * 4)               // bytes per index array tile (1 KB)
#define REL_TILES   (N_EDGES / TILE)         // 12500 tiles per relation
#define N_TILES     (N_REL * REL_TILES)      // 37500 total (exact division)

// Hardware fp32 atomic add at device (agent) scope -> global_atomic_add_f32.
__device__ __forceinline__ void atomic_add_f32(float* p, float v) {
    __hip_atomic_fetch_add(p, v, __ATOMIC_RELAXED, __HIP_MEMORY_SCOPE_AGENT);
}

// CDNA5 async copy: 16 bytes global -> LDS, no VGPR staging, tracked by ASYNCcnt.
// VDST operand = LDS byte address (low 32 bits of the generic shared pointer),
// VADDR = 64-bit global address, saddr = off.  (CDNA5 ISA)
__device__ __forceinline__ void async_copy_b128(uint32_t lds_addr, const void* gptr) {
    asm volatile("global_load_async_to_lds_b128 %0, %1, off"
                 :: "v"(lds_addr), "v"(gptr)
                 : "memory");
}

__device__ __forceinline__ void wait_asynccnt0() {
    asm volatile("s_wait_asynccnt 0x0" ::: "memory");
}

// Kernel 1: out = emb (float4 copy), deg[0..3N) = 0.
__global__ void k_init(const float* __restrict__ emb,
                       float* __restrict__ out,
                       float* __restrict__ deg) {
    const int tid = blockIdx.x * blockDim.x + threadIdx.x;
    const int n4 = (N_ITEMS * EMBED_DIM) / 4;          // 1,600,000
    if (tid < n4) {
        reinterpret_cast<float4*>(out)[tid] =
            reinterpret_cast<const float4*>(emb)[tid];
    }
    if (tid < N_REL * N_ITEMS) {
        deg[tid] = 0.0f;
    }
}

// Kernel 2: per-relation degree row-sums. One thread per edge (grid-stride).
__global__ void k_degree(const int* __restrict__ rows,
                         const float* __restrict__ vals,
                         float* __restrict__ deg) {
    const long long total  = (long long)N_REL * N_EDGES;
    const long long stride = (long long)gridDim.x * blockDim.x;
    for (long long e = (long long)blockIdx.x * blockDim.x + threadIdx.x;
         e < total; e += stride) {
        int rel = (e >= 2LL * N_EDGES) ? 2 : (e >= (long long)N_EDGES ? 1 : 0);
        atomic_add_f32(&deg[(size_t)rel * N_ITEMS + rows[e]], vals[e]);
    }
}

// Stage one 256-edge tile (rows | cols | vals, 3 KB) into an LDS buffer.
// Threads 0..191 each issue exactly one async b128 copy (192 * 16 B = 3 KB).
__device__ __forceinline__ void stage_tile(unsigned char* lds_buf,
                                           const int* __restrict__ rows,
                                           const int* __restrict__ cols,
                                           const float* __restrict__ vals,
                                           int tileStart, int t) {
    if (t < 192) {
        const int part = t >> 6;              // 0: rows, 1: cols, 2: vals
        const int off  = (t & 63) << 4;       // byte offset within 1 KB part
        const char* g = (part == 0) ? (const char*)(rows + tileStart)
                      : (part == 1) ? (const char*)(cols + tileStart)
                                    : (const char*)(vals + tileStart);
        const uint32_t lds_addr =
            (uint32_t)(uintptr_t)(lds_buf) + (uint32_t)(part * TILE_BYTES + off);
        async_copy_b128(lds_addr, g + off);
    }
}

// Kernel 3: weighted gather/scatter with async-staged, double-buffered edge tiles.
// 16 lanes per edge; each lane owns 4 consecutive floats (b128 gather + 4 atomics).
__global__ void __launch_bounds__(256)
k_scatter(const float* __restrict__ emb,
          const int* __restrict__ rows,
          const int* __restrict__ cols,
          const float* __restrict__ vals,
          const float* __restrict__ deg,
          float* __restrict__ out) {
    __shared__ __align__(16) unsigned char s_buf[2][3 * TILE_BYTES];   // 6 KB

    const int t   = threadIdx.x;
    const int grp = t >> 4;                   // edge group within pass [0,16)
    const int sub = t & 15;                   // lane within edge group

    int tile = blockIdx.x;
    if (tile >= N_TILES) return;              // uniform per block

    int buf = 0;
    stage_tile(s_buf[0], rows, cols, vals, tile * TILE, t);

    for (; tile < N_TILES; tile += gridDim.x) {
        wait_asynccnt0();                     // my staged copies have landed in LDS
        __syncthreads();                      // publish tile to all waves

        const int nextTile = tile + (int)gridDim.x;
        if (nextTile < N_TILES)               // overlap: stream next tile now
            stage_tile(s_buf[buf ^ 1], rows, cols, vals, nextTile * TILE, t);

        const int*   s_rows = (const int*)  (s_buf[buf]);
        const int*   s_cols = (const int*)  (s_buf[buf] + TILE_BYTES);
        const float* s_vals = (const float*)(s_buf[buf] + 2 * TILE_BYTES);

        const int rel = (tile >= 2 * REL_TILES) ? 2
                      : (tile >= REL_TILES)     ? 1 : 0;   // tiles never straddle relations
        const float* degR = deg + (size_t)rel * N_ITEMS;

        #pragma unroll 1
        for (int p = 0; p < TILE / 16; ++p) { // 16 edges per pass, 16 passes
            const int el = p * 16 + grp;

            // Prefetch next pass's gather fragment (global_prefetch_b8).
            if (p + 1 < TILE / 16) {
                const int ncol = s_cols[el + 16];
                __builtin_prefetch(emb + (size_t)ncol * EMBED_DIM + sub * 4, 0, 3);
            }

            const int   row = s_rows[el];     // LDS broadcast across the 16 lanes
            const int   col = s_cols[el];
            const float v   = s_vals[el];
            const float dg  = degR[row];
            const float w   = v * (1.0f / (float)N_REL) / fmaxf(dg, 1.0f);

            const float4 s = *reinterpret_cast<const float4*>(
                emb + (size_t)col * EMBED_DIM + sub * 4);
            float* o = out + (size_t)row * EMBED_DIM + sub * 4;
            atomic_add_f32(o + 0, s.x * w);
            atomic_add_f32(o + 1, s.y * w);
            atomic_add_f32(o + 2, s.z * w);
            atomic_add_f32(o + 3, s.w * w);
        }
        buf ^= 1;   // top-of-loop barrier protects reuse two iterations out
    }
}

extern "C" void kernel_launch(void* const* d_in, const int* in_sizes, int n_in,
                              void* d_out, int out_size, void* d_ws, size_t ws_size,
                              hipStream_t stream) {
    const float* emb  = (const float*)d_in[0];   // [N_ITEMS, 64] f32
    const int*   rows = (const int*)  d_in[1];   // [3, E] int
    const int*   cols = (const int*)  d_in[2];   // [3, E] int
    const float* vals = (const float*)d_in[3];   // [3, E] f32
    float*       out  = (float*)d_out;           // [N_ITEMS, 64] f32
    float*       deg  = (float*)d_ws;            // [3, N_ITEMS] f32 scratch (1.2 MB)

    {
        const int work = (N_ITEMS * EMBED_DIM) / 4;   // 1.6M >= 3*N
        const int blk  = 256;
        k_init<<<(work + blk - 1) / blk, blk, 0, stream>>>(emb, out, deg);
    }
    k_degree<<<8192, 256, 0, stream>>>(rows, vals, deg);
    // 4096 blocks x ~9 tiles each: enough waves to saturate, deep double-buffer pipeline.
    k_scatter<<<4096, 256, 0, stream>>>(emb, rows, cols, vals, deg, out);
}